// GraphConv_48687749268140
// MI455X (gfx1250) — compile-verified
//
#include <hip/hip_runtime.h>
#include <hip/hip_bf16.h>

// ---------------------------------------------------------------------------
// GraphConv: Lnorm = I - D^-1/2 A D^-1/2 ; eigh(Lnorm) -> U ;
// ret = U @ MLP(U^T @ F)   (MLP = relu(x W0^T + b0) W1^T + b1)
// Eigh via parallel cyclic Jacobi (1 workgroup / batch matrix).
// GEMMs: V_WMMA_F32_16X16X4_F32, 64x64 register blocking per wave (4x4 tiles),
// strided operand panels staged to LDS with GLOBAL_LOAD_ASYNC_TO_LDS_B128
// (ASYNCcnt protocol), streaming operands prefetched with global_prefetch_b8.
// ---------------------------------------------------------------------------

constexpr int BATCH  = 32;
constexpr int N      = 1024;
constexpr int INDIM  = 256;
constexpr int HID    = 512;
constexpr int OUTDIM = 256;
constexpr int NSWEEPS = 12;

typedef float v2f __attribute__((ext_vector_type(2)));
typedef float v8f __attribute__((ext_vector_type(8)));

__device__ __forceinline__ void async_ld_b128(unsigned lds_off, const float* gptr) {
  // copies 16 bytes from the global address in gptr to the LDS address in lds_off; tracked by ASYNCcnt
  asm volatile("global_load_async_to_lds_b128 %0, %1, off"
               :: "v"(lds_off), "v"(gptr) : "memory");
}
__device__ __forceinline__ void wait_async0() {
  asm volatile("s_wait_asynccnt 0x0" ::: "memory");
}

// ---------------- degree -> d^{-1/2} ---------------------------------------
__global__ void k_deg(const float* __restrict__ A, float* __restrict__ dis) {
  const int row = blockIdx.x & (N - 1);
  const int b   = blockIdx.x >> 10;
  const float* Ar = A + ((size_t)b * N + row) * N;
  float sum = 0.f;
  for (int j = threadIdx.x; j < N; j += 256) sum += Ar[j];
  __shared__ float red[256];
  red[threadIdx.x] = sum; __syncthreads();
  for (int off = 128; off > 0; off >>= 1) {
    if (threadIdx.x < off) red[threadIdx.x] += red[threadIdx.x + off];
    __syncthreads();
  }
  if (threadIdx.x == 0) dis[(size_t)b * N + row] = rsqrtf(red[0]);
}

// ---------------- M = I - D^-1/2 A D^-1/2 ; U = I --------------------------
__global__ void k_lap(const float* __restrict__ A, const float* __restrict__ dis,
                      float* __restrict__ M, float* __restrict__ U) {
  size_t idx = (size_t)blockIdx.x * 256 + threadIdx.x;   // < BATCH*N*N
  int c = (int)(idx & (N - 1));
  int r = (int)((idx >> 10) & (N - 1));
  int b = (int)(idx >> 20);
  float v = -dis[(size_t)b * N + r] * A[idx] * dis[(size_t)b * N + c];
  float diag = (r == c) ? 1.f : 0.f;
  M[idx] = v + diag;
  U[idx] = diag;
}

// ---------------- parallel cyclic Jacobi eigensolver -----------------------
__global__ __launch_bounds__(1024) void k_jacobi(float* __restrict__ Mg,
                                                 float* __restrict__ Ug) {
  const int b = blockIdx.x;
  float* M = Mg + (size_t)b * N * N;
  float* U = Ug + (size_t)b * N * N;
  const int tid = threadIdx.x;
  const int m = N - 1;                       // 1023
  __shared__ float sc[N / 2], ss[N / 2];
  __shared__ int   sp[N / 2], sq[N / 2];

  for (int sweep = 0; sweep < NSWEEPS; ++sweep) {
    for (int r = 0; r < m; ++r) {
      // --- rotation angles from current snapshot (512 disjoint pairs) ---
      if (tid < N / 2) {
        int p, q;
        if (tid == 0) { p = m; q = r % m; }
        else {
          p = (r + tid) % m;
          q = (r - tid + m) % m;
        }
        if (p > q) { int t = p; p = q; q = t; }
        sp[tid] = p; sq[tid] = q;
        float app = M[(size_t)p * N + p];
        float aqq = M[(size_t)q * N + q];
        float apq = M[(size_t)p * N + q];
        float c = 1.f, s = 0.f;
        if (fabsf(apq) > 1e-30f) {
          float tau = (aqq - app) / (2.f * apq);
          float t   = copysignf(1.f, tau) / (fabsf(tau) + sqrtf(1.f + tau * tau));
          c = rsqrtf(1.f + t * t);
          s = t * c;
        }
        sc[tid] = c; ss[tid] = s;
      }
      __syncthreads();
      // --- column phase: M <- M J ; U <- U J (disjoint column pairs) ---
      for (int w = tid; w < (N / 2) * N; w += 1024) {
        int pr  = w & (N / 2 - 1);
        int row = w >> 9;
        int p = sp[pr], q = sq[pr];
        float c = sc[pr], s = ss[pr];
        size_t ip = (size_t)row * N + p, iq = (size_t)row * N + q;
        float mp = M[ip], mq = M[iq];
        M[ip] = c * mp - s * mq;
        M[iq] = s * mp + c * mq;
        float up = U[ip], uq = U[iq];
        U[ip] = c * up - s * uq;
        U[iq] = s * up + c * uq;
      }
      __syncthreads();
      // --- row phase: M <- J^T M (disjoint row pairs, coalesced) ---
      for (int w = tid; w < (N / 2) * N; w += 1024) {
        int col = w & (N - 1);
        int pr  = w >> 10;
        int p = sp[pr], q = sq[pr];
        float c = sc[pr], s = ss[pr];
        size_t ip = (size_t)p * N + col, iq = (size_t)q * N + col;
        float mp = M[ip], mq = M[iq];
        M[ip] = c * mp - s * mq;
        M[iq] = s * mp + c * mq;
      }
      __syncthreads();
    }
  }
}

// ---------------- ascending sort of eigenvalues -> permutation -------------
__global__ __launch_bounds__(1024) void k_sortperm(const float* __restrict__ M,
                                                   int* __restrict__ perm) {
  const int b = blockIdx.x;
  __shared__ float lam[N];
  const int j = threadIdx.x;
  lam[j] = M[(size_t)b * N * N + (size_t)j * N + j];
  __syncthreads();
  float l = lam[j];
  int rank = 0;
  for (int k = 0; k < N; ++k) {
    float lk = lam[k];
    rank += (lk < l) || (lk == l && k < j);
  }
  perm[b * N + rank] = j;
}

// ---------------- deterministic column sign convention ---------------------
__global__ void k_colsign(const float* __restrict__ U, const int* __restrict__ perm,
                          float* __restrict__ sgn) {
  const int c = blockIdx.x & (N - 1);
  const int b = blockIdx.x >> 10;
  const int jc = perm[b * N + c];
  const float* col = U + (size_t)b * N * N + jc;
  float bestv = 0.f; int besti = N;
  for (int i = threadIdx.x; i < N; i += 128) {
    float v = col[(size_t)i * N];
    if (fabsf(v) > fabsf(bestv) || (fabsf(v) == fabsf(bestv) && i < besti)) {
      bestv = v; besti = i;
    }
  }
  __shared__ float rv[128]; __shared__ int ri[128];
  rv[threadIdx.x] = bestv; ri[threadIdx.x] = besti; __syncthreads();
  for (int off = 64; off > 0; off >>= 1) {
    if (threadIdx.x < off) {
      float v2 = rv[threadIdx.x + off]; int i2 = ri[threadIdx.x + off];
      if (fabsf(v2) > fabsf(rv[threadIdx.x]) ||
          (fabsf(v2) == fabsf(rv[threadIdx.x]) && i2 < ri[threadIdx.x])) {
        rv[threadIdx.x] = v2; ri[threadIdx.x] = i2;
      }
    }
    __syncthreads();
  }
  if (threadIdx.x == 0) sgn[b * N + c] = (rv[0] < 0.f) ? -1.f : 1.f;
}

// ---------------- Us[:, r] = sgn[r] * U[:, perm[r]] ------------------------
__global__ void k_gather(const float* __restrict__ U, const int* __restrict__ perm,
                         const float* __restrict__ sgn, float* __restrict__ Us) {
  size_t idx = (size_t)blockIdx.x * 256 + threadIdx.x;
  int c = (int)(idx & (N - 1));
  int i = (int)((idx >> 10) & (N - 1));
  int b = (int)(idx >> 20);
  int jc = perm[b * N + c];
  Us[idx] = sgn[b * N + c] * U[(size_t)b * N * N + (size_t)i * N + jc];
}

// ---------------- WMMA GEMMs (fp32 16x16x4, 64x64 wave macro-tile) ---------
// ISA VGPR layouts: A 16x4 (M=lane%16, K=j+2*(lane/16)),
// B 4x16 (K=j+2*(lane/16), Ncol=lane%16), C/D 16x16 (M=j+8*(lane/16), Ncol=lane%16).

#define ZERO_ACC8 {0.f, 0.f, 0.f, 0.f, 0.f, 0.f, 0.f, 0.f}
#define INIT_ACC44(acc)                                     \
  v8f acc[4][4];                                            \
  _Pragma("unroll") for (int i = 0; i < 4; ++i)             \
  _Pragma("unroll") for (int j = 0; j < 4; ++j)             \
    acc[i][j] = (v8f)ZERO_ACC8;

// agg[b][n][d] = sum_m Us[b][m][n] * F[b][m][d]
// A-panel (columns of Us, shared by 4 waves) staged to LDS via async b128.
__global__ __launch_bounds__(256) void k_gemm1(const float* __restrict__ Us,
                                               const float* __restrict__ F,
                                               float* __restrict__ agg) {
  const int b      = blockIdx.y;
  const int w      = threadIdx.x >> 5;            // wave in block: 0..7
  const int bnBase = blockIdx.x << 7;             // 8 blocks cover n in 128s
  const int bn     = bnBase + ((w >> 2) << 6);    // 2 n-groups of 64
  const int bd     = (w & 3) << 6;                // 4 d-groups of 64
  const int lane   = threadIdx.x & 31;
  const int lhi    = lane >> 4, llo = lane & 15;
  const float* Ub = Us + (size_t)b * N * N;
  const float* Fb = F + (size_t)b * N * INDIM;

  __shared__ float ldsA[32 * 128];                // 16 KB K-panel
  const unsigned ldsbase = (unsigned)(uintptr_t)(void*)ldsA;
  const float* As = ldsA + (bn - bnBase);

  INIT_ACC44(acc);
  for (int kp = 0; kp < N; kp += 32) {
    // ---- stage A panel rows [kp,kp+32) x cols [bnBase,bnBase+128) ----
#pragma unroll
    for (int e = 0; e < 4; ++e) {
      int idx  = threadIdx.x + 256 * e;           // 0..1023 x 4 floats
      int row  = idx >> 5;                        // 0..31
      int col4 = (idx & 31) << 2;                 // 0,4,..,124
      async_ld_b128(ldsbase + (unsigned)((row * 128 + col4) * 4),
                    Ub + (size_t)(kp + row) * N + bnBase + col4);
    }
    if (kp + 32 < N)
      __builtin_prefetch(Fb + (size_t)(kp + 32 + (threadIdx.x >> 5)) * INDIM +
                             (threadIdx.x & 31) * 8, 0, 1);
    wait_async0();                                // own wave's ASYNCcnt
    __syncthreads();                              // panel visible to all waves
    // ---- consume panel: 8 K-steps of 4 ----
    for (int kk = 0; kk < 32; kk += 4) {
      const int ka = kk + 2 * lhi;
      v2f av[4], bv[4];
#pragma unroll
      for (int i = 0; i < 4; ++i) {
        av[i].x = As[(ka + 0) * 128 + 16 * i + llo];
        av[i].y = As[(ka + 1) * 128 + 16 * i + llo];
        bv[i].x = Fb[(size_t)(kp + ka + 0) * INDIM + bd + 16 * i + llo];
        bv[i].y = Fb[(size_t)(kp + ka + 1) * INDIM + bd + 16 * i + llo];
      }
#pragma unroll
      for (int i = 0; i < 4; ++i)
#pragma unroll
        for (int j = 0; j < 4; ++j)
          acc[i][j] = __builtin_amdgcn_wmma_f32_16x16x4_f32(
              false, av[i], false, bv[j], (short)0, acc[i][j], false, false);
    }
    __syncthreads();                              // before panel overwrite
  }
  float* Cb = agg + (size_t)b * N * INDIM;
#pragma unroll
  for (int i = 0; i < 4; ++i)
#pragma unroll
    for (int j = 0; j < 4; ++j)
#pragma unroll
      for (int r = 0; r < 8; ++r)
        Cb[(size_t)(bn + 16 * i + r + 8 * lhi) * INDIM + bd + 16 * j + llo] =
            acc[i][j][r];
}

// h[b][n][o] = relu(sum_k agg[b][n][k] * W0[o][k] + b0[o])
__global__ __launch_bounds__(256) void k_gemm2(const float* __restrict__ agg,
                                               const float* __restrict__ W0,
                                               const float* __restrict__ b0,
                                               float* __restrict__ h) {
  const int b    = blockIdx.y;
  const int wave = (blockIdx.x * 256 + threadIdx.x) >> 5;  // 0..127
  const int bn   = (wave >> 3) << 6;                       // 16 n-blocks of 64
  const int bh   = (wave & 7) << 6;                        // 8 hid-blocks of 64
  const int lane = threadIdx.x & 31;
  const int lhi  = lane >> 4, llo = lane & 15;
  const float* Ab = agg + (size_t)b * N * INDIM;
  INIT_ACC44(acc);
  for (int k0 = 0; k0 < INDIM; k0 += 4) {
    const int ka = k0 + 2 * lhi;
    v2f av[4], bv[4];
#pragma unroll
    for (int i = 0; i < 4; ++i) {
      av[i] = *(const v2f*)&Ab[(size_t)(bn + 16 * i + llo) * INDIM + ka];
      bv[i] = *(const v2f*)&W0[(size_t)(bh + 16 * i + llo) * INDIM + ka];
    }
#pragma unroll
    for (int i = 0; i < 4; ++i)
#pragma unroll
      for (int j = 0; j < 4; ++j)
        acc[i][j] = __builtin_amdgcn_wmma_f32_16x16x4_f32(
            false, av[i], false, bv[j], (short)0, acc[i][j], false, false);
  }
  float* Cb = h + (size_t)b * N * HID;
#pragma unroll
  for (int j = 0; j < 4; ++j) {
    const float bias = b0[bh + 16 * j + llo];
#pragma unroll
    for (int i = 0; i < 4; ++i)
#pragma unroll
      for (int r = 0; r < 8; ++r) {
        float v = acc[i][j][r] + bias;
        Cb[(size_t)(bn + 16 * i + r + 8 * lhi) * HID + bh + 16 * j + llo] =
            v > 0.f ? v : 0.f;
      }
  }
}

// out[b][n][o] = sum_k h[b][n][k] * W1[o][k] + b1[o]
__global__ __launch_bounds__(256) void k_gemm3(const float* __restrict__ h,
                                               const float* __restrict__ W1,
                                               const float* __restrict__ b1,
                                               float* __restrict__ outp) {
  const int b    = blockIdx.y;
  const int wave = (blockIdx.x * 256 + threadIdx.x) >> 5;  // 0..63
  const int bn   = (wave >> 2) << 6;                       // 16 n-blocks of 64
  const int bo   = (wave & 3) << 6;                        // 4 out-blocks of 64
  const int lane = threadIdx.x & 31;
  const int lhi  = lane >> 4, llo = lane & 15;
  const float* Ab = h + (size_t)b * N * HID;
  INIT_ACC44(acc);
  for (int k0 = 0; k0 < HID; k0 += 4) {
    const int ka = k0 + 2 * lhi;
    v2f av[4], bv[4];
#pragma unroll
    for (int i = 0; i < 4; ++i) {
      av[i] = *(const v2f*)&Ab[(size_t)(bn + 16 * i + llo) * HID + ka];
      bv[i] = *(const v2f*)&W1[(size_t)(bo + 16 * i + llo) * HID + ka];
    }
#pragma unroll
    for (int i = 0; i < 4; ++i)
#pragma unroll
      for (int j = 0; j < 4; ++j)
        acc[i][j] = __builtin_amdgcn_wmma_f32_16x16x4_f32(
            false, av[i], false, bv[j], (short)0, acc[i][j], false, false);
  }
  float* Cb = outp + (size_t)b * N * OUTDIM;
#pragma unroll
  for (int j = 0; j < 4; ++j) {
    const float bias = b1[bo + 16 * j + llo];
#pragma unroll
    for (int i = 0; i < 4; ++i)
#pragma unroll
      for (int r = 0; r < 8; ++r)
        Cb[(size_t)(bn + 16 * i + r + 8 * lhi) * OUTDIM + bo + 16 * j + llo] =
            acc[i][j][r] + bias;
  }
}

// ret[b][n][o] = sum_m Us[b][n][m] * out[b][m][o]
// B-panel (rows of out, strided per K, shared by 2 waves) staged via async b128.
__global__ __launch_bounds__(256) void k_gemm4(const float* __restrict__ Us,
                                               const float* __restrict__ outp,
                                               float* __restrict__ ret) {
  const int b      = blockIdx.y;
  const int w      = threadIdx.x >> 5;            // wave in block: 0..7
  const int bnBase = blockIdx.x << 7;             // 8 blocks cover n in 128s
  const int bn     = bnBase + ((w >> 2) << 6);
  const int bo     = (w & 3) << 6;
  const int lane   = threadIdx.x & 31;
  const int lhi    = lane >> 4, llo = lane & 15;
  const float* Ub = Us + (size_t)b * N * N;
  const float* Ob = outp + (size_t)b * N * OUTDIM;

  __shared__ float ldsB[32 * 256];                // 32 KB K-panel of out
  const unsigned ldsbase = (unsigned)(uintptr_t)(void*)ldsB;

  INIT_ACC44(acc);
  for (int kp = 0; kp < N; kp += 32) {
    // ---- stage B panel rows [kp,kp+32) x all 256 cols ----
#pragma unroll
    for (int e = 0; e < 8; ++e) {
      int idx  = threadIdx.x + 256 * e;           // 0..2047 x 4 floats
      int row  = idx >> 6;                        // 0..31
      int col4 = (idx & 63) << 2;                 // 0,4,..,252
      async_ld_b128(ldsbase + (unsigned)((row * 256 + col4) * 4),
                    Ob + (size_t)(kp + row) * OUTDIM + col4);
    }
    wait_async0();
    __syncthreads();
    // ---- consume panel ----
    for (int kk = 0; kk < 32; kk += 4) {
      const int ka = kk + 2 * lhi;
      v2f av[4], bv[4];
#pragma unroll
      for (int i = 0; i < 4; ++i) {
        av[i] = *(const v2f*)&Ub[(size_t)(bn + 16 * i + llo) * N + kp + ka];
        bv[i].x = ldsB[(ka + 0) * 256 + bo + 16 * i + llo];
        bv[i].y = ldsB[(ka + 1) * 256 + bo + 16 * i + llo];
      }
#pragma unroll
      for (int i = 0; i < 4; ++i)
#pragma unroll
        for (int j = 0; j < 4; ++j)
          acc[i][j] = __builtin_amdgcn_wmma_f32_16x16x4_f32(
              false, av[i], false, bv[j], (short)0, acc[i][j], false, false);
    }
    __syncthreads();
  }
  float* Cb = ret + (size_t)b * N * OUTDIM;
#pragma unroll
  for (int i = 0; i < 4; ++i)
#pragma unroll
    for (int j = 0; j < 4; ++j)
#pragma unroll
      for (int r = 0; r < 8; ++r)
        Cb[(size_t)(bn + 16 * i + r + 8 * lhi) * OUTDIM + bo + 16 * j + llo] =
            acc[i][j][r];
}

// ---------------------------------------------------------------------------
extern "C" void kernel_launch(void* const* d_in, const int* in_sizes, int n_in,
                              void* d_out, int out_size, void* d_ws, size_t ws_size,
                              hipStream_t stream) {
  (void)in_sizes; (void)n_in; (void)out_size; (void)ws_size;
  const float* F  = (const float*)d_in[0];
  const float* A  = (const float*)d_in[1];
  const float* W0 = (const float*)d_in[2];
  const float* b0 = (const float*)d_in[3];
  const float* W1 = (const float*)d_in[4];
  const float* b1 = (const float*)d_in[5];
  float* ret = (float*)d_out;

  // workspace layout (floats): [M/Us: B*N*N][U: B*N*N][agg|out][h][dis][sgn][perm]
  float* wsf = (float*)d_ws;
  const size_t NNs = (size_t)BATCH * N * N;
  float* Ms   = wsf;                               // Laplacian -> sorted U
  float* Uu   = Ms + NNs;                          // Jacobi eigenvectors
  float* agg  = Uu + NNs;                          // agg, later reused as 'out'
  float* hbuf = agg + (size_t)BATCH * N * INDIM;
  float* dis  = hbuf + (size_t)BATCH * N * HID;
  float* sgn  = dis + (size_t)BATCH * N;
  int*   perm = (int*)(sgn + (size_t)BATCH * N);

  const int elemBlocks = (int)(NNs / 256);         // 131072

  k_deg     <<<BATCH * N, 256, 0, stream>>>(A, dis);
  k_lap     <<<elemBlocks, 256, 0, stream>>>(A, dis, Ms, Uu);
  k_jacobi  <<<BATCH, 1024, 0, stream>>>(Ms, Uu);
  k_sortperm<<<BATCH, 1024, 0, stream>>>(Ms, perm);
  k_colsign <<<BATCH * N, 128, 0, stream>>>(Uu, perm, sgn);
  k_gather  <<<elemBlocks, 256, 0, stream>>>(Uu, perm, sgn, Ms);  // Ms := sorted/signed U
  k_gemm1   <<<dim3(8,  BATCH), 256, 0, stream>>>(Ms, F, agg);
  k_gemm2   <<<dim3(16, BATCH), 256, 0, stream>>>(agg, W0, b0, hbuf);
  k_gemm3   <<<dim3(8,  BATCH), 256, 0, stream>>>(hbuf, W1, b1, agg);  // agg := out
  k_gemm4   <<<dim3(8,  BATCH), 256, 0, stream>>>(Ms, agg, ret);
}